// BertCRF_8134668059192
// MI455X (gfx1250) — compile-verified
//
#include <hip/hip_runtime.h>
#include <hip/hip_bf16.h>
#include <math.h>

// ---------------------------------------------------------------------------
// BertCRF on MI455X (gfx1250, wave32)
//   Kernel 1: fused embedding-gather + (16384x768)@(768x17) GEMM via
//             V_WMMA_F32_16X16X4_F32 + bias + log_softmax  -> logits in d_ws
//             (W staged in LDS so VMEM path carries only the 50 MB gather)
//   Kernel 2: CRF forward DP, 1 exp + 1 log per lane per step; all cross-lane
//             reductions via single-instruction ds_swizzle_b32 xor patterns.
// ---------------------------------------------------------------------------

typedef float v2f __attribute__((ext_vector_type(2)));
typedef float v8f __attribute__((ext_vector_type(8)));

#define EMBED 768
#define NTAGS 17
#define BATCH 32
#define SEQ   512
#define ROWS  (BATCH * SEQ)   // 16384

// ds_swizzle_b32, group-of-32 mode: and_mask=0x1F, or_mask=0, xor_mask=X
#define SWZ_XOR_I(v, X) __builtin_amdgcn_ds_swizzle((v), (0x1F | ((X) << 10)))
#define SWZ_XOR_F(v, X) __int_as_float(SWZ_XOR_I(__float_as_int(v), (X)))

// ---------------- Kernel 1: gather + WMMA GEMM + log_softmax ----------------
// One wave per 16-row tile of the (16384 x 17) logits matrix.
// blockDim = 256 (8 waves/block), grid = 16384 / (16*8) = 128 blocks.
__global__ __launch_bounds__(256)
void embed_logits_kernel(const int* __restrict__ words,
                         const float* __restrict__ table,
                         const float* __restrict__ W,      // (768, 17) row-major
                         const float* __restrict__ bias,   // (17,)
                         float* __restrict__ logits)       // (16384, 17)
{
    __shared__ float sW[EMBED * NTAGS];   // 52224 B: whole weight matrix

    const int tid = threadIdx.x;
    for (int i = tid; i < EMBED * NTAGS; i += 256) sW[i] = W[i];
    __syncthreads();

    const int lane = tid & 31;
    const int wave = tid >> 5;
    const int tile = blockIdx.x * 8 + wave;   // 16 rows per tile
    const int half = lane >> 4;               // K sub-pair selector
    const int sub  = lane & 15;               // A-row / B-column within tile

    // A fragment: this lane streams row (tile*16+sub), K = k0 + 2*half + {0,1}
    const int row  = tile * 16 + sub;
    const int word = words[row];
    const float* __restrict__ arow = table + (size_t)word * EMBED + 2 * half;
    // B fragment from LDS: sW[(k0 + 2*half + {0,1})*17 + sub]
    const int wb   = 2 * half * NTAGS + sub;
    const int wb16 = 2 * half * NTAGS + 16;

    v8f   c = {};          // 16x16 fp32 accumulator (tags 0..15)
    float acc16 = 0.0f;    // tag 16 partial dot

    #pragma unroll 4
    for (int k0 = 0; k0 < EMBED; k0 += 4) {
        const float2 a2 = *(const float2*)(arow + k0);   // HBM gather stream
        v2f a; a.x = a2.x; a.y = a2.y;
        v2f bf; bf.x = sW[wb + k0 * NTAGS]; bf.y = sW[wb + (k0 + 1) * NTAGS];
        c = __builtin_amdgcn_wmma_f32_16x16x4_f32(
                false, a, false, bf, (short)0, c, false, false);
        acc16 = fmaf(a2.x, sW[wb16 + k0 * NTAGS], acc16);
        acc16 = fmaf(a2.y, sW[wb16 + (k0 + 1) * NTAGS], acc16);
    }

    // Combine the two K-halves of tag-16 (lane L <-> L^16); add biases.
    acc16 += SWZ_XOR_F(acc16, 16);
    acc16 += bias[16];
    const float cbias = bias[sub];

    // C layout (ISA 7.12.2): VGPR r -> row r (lanes 0-15) / row r+8 (16-31),
    // column = lane&15.  log_softmax over 17 values per row.
    #pragma unroll
    for (int r = 0; r < 8; ++r) {
        float val = c[r] + cbias;
        // tag16 of rows r and r+8 (uniform lane index -> v_readlane)
        const float t16lo = __int_as_float(
            __builtin_amdgcn_readlane(__float_as_int(acc16), r));
        const float t16hi = __int_as_float(
            __builtin_amdgcn_readlane(__float_as_int(acc16), r + 8));
        const float v16 = half ? t16hi : t16lo;   // uniform per half-wave

        float m = val;                            // row max over 16 lanes
        m = fmaxf(m, SWZ_XOR_F(m, 8));
        m = fmaxf(m, SWZ_XOR_F(m, 4));
        m = fmaxf(m, SWZ_XOR_F(m, 2));
        m = fmaxf(m, SWZ_XOR_F(m, 1));
        m = fmaxf(m, v16);

        float e = __expf(val - m);
        e += SWZ_XOR_F(e, 8);
        e += SWZ_XOR_F(e, 4);
        e += SWZ_XOR_F(e, 2);
        e += SWZ_XOR_F(e, 1);
        const float lse = m + __logf(e + __expf(v16 - m));

        const int orow = tile * 16 + r + 8 * half;
        logits[orow * NTAGS + sub] = val - lse;
        if (sub == 0) logits[orow * NTAGS + 16] = v16 - lse;
    }
}

// ---------------- Kernel 2: CRF forward DP + gold score ---------------------
// One wave (32 threads) per batch sample; lane j (<17) owns alpha[j].
// new[j] = emit[j] + m + log(sum_i exp(alpha[i]-m) * exp(trans[i][j]))
__global__ __launch_bounds__(32)
void crf_kernel(const int* __restrict__ words,
                const int* __restrict__ target,
                const float* __restrict__ logits,   // (16384, 17) log-probs
                const float* __restrict__ trans,    // (17, 17)
                const float* __restrict__ startv,   // (17,)
                const float* __restrict__ endv,     // (17,)
                float* __restrict__ out)            // (32,)
{
    const int b    = blockIdx.x;
    const int lane = threadIdx.x;          // 0..31
    const int base = b * SEQ;
    const bool active = lane < NTAGS;
    const int  j = active ? lane : (NTAGS - 1);

    __shared__ __align__(16) float sEA[32];

    // exp(trans) column in registers (constant across all 511 steps).
    float et[NTAGS];
    #pragma unroll
    for (int i = 0; i < NTAGS; ++i) et[i] = __expf(trans[i * NTAGS + j]);

    // ---- gold path score (all 32 lanes, strided over time) ----
    float acc = 0.0f; int cnt = 0;
    for (int t = lane; t < SEQ; t += 32) {
        const int wd = words[base + t];
        const int tg = target[base + t];
        if (wd != 0) {
            acc += logits[(base + t) * NTAGS + tg];
            if (t > 0) acc += trans[target[base + t - 1] * NTAGS + tg];
            cnt += 1;
        }
    }
    acc += SWZ_XOR_F(acc, 16); cnt += SWZ_XOR_I(cnt, 16);
    acc += SWZ_XOR_F(acc, 8);  cnt += SWZ_XOR_I(cnt, 8);
    acc += SWZ_XOR_F(acc, 4);  cnt += SWZ_XOR_I(cnt, 4);
    acc += SWZ_XOR_F(acc, 2);  cnt += SWZ_XOR_I(cnt, 2);
    acc += SWZ_XOR_F(acc, 1);  cnt += SWZ_XOR_I(cnt, 1);
    const int last = (cnt > 0) ? (cnt - 1) : 0;
    const float score = acc + startv[target[base]] + endv[target[base + last]];

    // ---- forward algorithm ----
    float alpha = active ? (logits[base * NTAGS + lane] + startv[lane]) : -INFINITY;

    for (int t0 = 0; t0 < SEQ; t0 += 32) {
        // one uniform load + ballot covers the mask for 32 steps
        const unsigned long long mbits =
            __ballot(words[base + t0 + lane] != 0);
        const int tstart = (t0 == 0) ? 1 : t0;

        for (int t = tstart; t < t0 + 32; ++t) {
            const float emit = logits[(size_t)(base + t) * NTAGS + j];

            float m = alpha;                       // max_i alpha[i]
            m = fmaxf(m, SWZ_XOR_F(m, 16));
            m = fmaxf(m, SWZ_XOR_F(m, 8));
            m = fmaxf(m, SWZ_XOR_F(m, 4));
            m = fmaxf(m, SWZ_XOR_F(m, 2));
            m = fmaxf(m, SWZ_XOR_F(m, 1));

            sEA[lane] = active ? __expf(alpha - m) : 0.0f;
            __syncthreads();
            const float4 q0 = *(const float4*)(sEA + 0);
            const float4 q1 = *(const float4*)(sEA + 4);
            const float4 q2 = *(const float4*)(sEA + 8);
            const float4 q3 = *(const float4*)(sEA + 12);
            const float qe = sEA[16];
            __syncthreads();

            float s0 = q0.x * et[0];
            float s1 = q0.y * et[1];
            s0 = fmaf(q0.z, et[2],  s0); s1 = fmaf(q0.w, et[3],  s1);
            s0 = fmaf(q1.x, et[4],  s0); s1 = fmaf(q1.y, et[5],  s1);
            s0 = fmaf(q1.z, et[6],  s0); s1 = fmaf(q1.w, et[7],  s1);
            s0 = fmaf(q2.x, et[8],  s0); s1 = fmaf(q2.y, et[9],  s1);
            s0 = fmaf(q2.z, et[10], s0); s1 = fmaf(q2.w, et[11], s1);
            s0 = fmaf(q3.x, et[12], s0); s1 = fmaf(q3.y, et[13], s1);
            s0 = fmaf(q3.z, et[14], s0); s1 = fmaf(q3.w, et[15], s1);
            s0 = fmaf(qe,   et[16], s0);
            const float s = s0 + s1;

            const float na = emit + m + __logf(s);
            const bool mk = (mbits >> (t - t0)) & 1ull;
            if (active && mk) alpha = na;
        }
    }

    // ---- normalizer = LSE(alpha + end) ----
    float fa = active ? (alpha + endv[lane]) : -INFINITY;
    float m2 = fa;
    m2 = fmaxf(m2, SWZ_XOR_F(m2, 16));
    m2 = fmaxf(m2, SWZ_XOR_F(m2, 8));
    m2 = fmaxf(m2, SWZ_XOR_F(m2, 4));
    m2 = fmaxf(m2, SWZ_XOR_F(m2, 2));
    m2 = fmaxf(m2, SWZ_XOR_F(m2, 1));
    float e2 = active ? __expf(fa - m2) : 0.0f;
    e2 += SWZ_XOR_F(e2, 16);
    e2 += SWZ_XOR_F(e2, 8);
    e2 += SWZ_XOR_F(e2, 4);
    e2 += SWZ_XOR_F(e2, 2);
    e2 += SWZ_XOR_F(e2, 1);
    const float normalizer = m2 + __logf(e2);

    if (lane == 0) out[b] = normalizer - score;
}

// ---------------------------- host launcher ---------------------------------
extern "C" void kernel_launch(void* const* d_in, const int* in_sizes, int n_in,
                              void* d_out, int out_size, void* d_ws, size_t ws_size,
                              hipStream_t stream) {
    // setup_inputs order: words, target, embed_table, W, b, trans, start, end
    const int*   words  = (const int*)  d_in[0];
    const int*   target = (const int*)  d_in[1];
    const float* table  = (const float*)d_in[2];
    const float* W      = (const float*)d_in[3];
    const float* bias   = (const float*)d_in[4];
    const float* trans  = (const float*)d_in[5];
    const float* startv = (const float*)d_in[6];
    const float* endv   = (const float*)d_in[7];
    float*       out    = (float*)d_out;
    float*       logits = (float*)d_ws;            // 16384*17 floats ~= 1.1 MB

    (void)in_sizes; (void)n_in; (void)out_size; (void)ws_size;

    // 1024 tiles of 16 rows; 8 waves/block -> 128 blocks of 256 threads.
    embed_logits_kernel<<<ROWS / (16 * 8), 256, 0, stream>>>(
        words, table, W, bias, logits);

    // one wave per batch sample
    crf_kernel<<<BATCH, 32, 0, stream>>>(
        words, target, logits, trans, startv, endv, out);
}